// GNN_72825465471510
// MI455X (gfx1250) — compile-verified
//
#include <hip/hip_runtime.h>
#include <hip/hip_bf16.h>
#include <math.h>

typedef float v2f __attribute__((ext_vector_type(2)));
typedef float v8f __attribute__((ext_vector_type(8)));

#define GF_RELU      1
#define GF_ADD_RESID 2
#define GF_RELU_ADD  4
#define GF_RESID_VEC 8

// ---------------------------------------------------------------------------
// WMMA f32 GEMM: out[M,NC] = A[M,K(lda)] * W[NC,K]^T (+bias) (+epilogue)
// K, FLAGS, HAS_BIAS compile-time -> fully unrolled straight-line code,
// unconditional 64-bit loads. One wave per 16-row tile; A preloaded once.
// A 16x4 layout: lane L: row = L%16, VGPR j -> k = 2*(L/16)+j
// B 4x16 layout: lane L: col = L%16, VGPR j -> k = 2*(L/16)+j
// D 16x16:       VGPR r: row = r + 8*(L/16), col = L%16
// Requires M % 16 == 0 (true for all call sites: 512000, 20480).
// ---------------------------------------------------------------------------
template<int K, int FLAGS, bool HAS_BIAS>
__global__ void gemm_wmma_k(const float* __restrict__ A, int lda,
                            const float* __restrict__ W,
                            const float* __restrict__ bias,
                            const float* __restrict__ resid, int ldr,
                            float* __restrict__ out, int ldo,
                            int M, int NC)
{
    const int wave = blockIdx.x * (blockDim.x >> 5) + (threadIdx.x >> 5);
    const int lane = threadIdx.x & 31;
    const int m0 = wave * 16;
    if (m0 >= M) return;
    const int half = lane >> 4;
    const int l16  = lane & 15;
    const float* Arow = A + (size_t)(m0 + l16) * lda + 2 * half;

    v2f areg[K / 4];
    #pragma unroll
    for (int kk = 0; kk < K / 4; ++kk)
        areg[kk] = *(const v2f*)(Arow + 4 * kk);

    for (int n0 = 0; n0 < NC; n0 += 16) {
        const float* Wrow = W + (size_t)(n0 + l16) * K + 2 * half;
        v8f acc = {0.f, 0.f, 0.f, 0.f, 0.f, 0.f, 0.f, 0.f};
        #pragma unroll
        for (int kk = 0; kk < K / 4; ++kk) {
            v2f b = *(const v2f*)(Wrow + 4 * kk);
            acc = __builtin_amdgcn_wmma_f32_16x16x4_f32(false, areg[kk], false, b,
                                                        (short)0, acc, false, false);
        }
        const float bc = HAS_BIAS ? bias[n0 + l16] : 0.f;
        float rv[8];
        if (FLAGS & (GF_ADD_RESID | GF_RELU_ADD)) {
            if (FLAGS & GF_RESID_VEC) {
                const float r0 = resid[n0 + l16];
                #pragma unroll
                for (int r = 0; r < 8; ++r) rv[r] = r0;
            } else {
                #pragma unroll
                for (int r = 0; r < 8; ++r)
                    rv[r] = resid[(size_t)(m0 + r + 8 * half) * ldr + n0 + l16];
            }
        }
        #pragma unroll
        for (int r = 0; r < 8; ++r) {
            const int m = m0 + r + 8 * half;
            float v = acc[r] + bc;
            if (FLAGS & GF_RELU) v = fmaxf(v, 0.f);
            if (FLAGS & GF_RELU_ADD) v = fmaxf(v, 0.f) + rv[r];
            else if (FLAGS & GF_ADD_RESID) v = v + rv[r];
            out[(size_t)m * ldo + n0 + l16] = v;
        }
    }
}

template<int K, int FLAGS, bool HAS_BIAS>
static inline void launch_gemm(const float* A, int lda, const float* W,
                               const float* bias, const float* resid, int ldr,
                               float* out, int ldo, int M, int NC, hipStream_t s)
{
    int blocks = (((M + 15) >> 4) + 7) >> 3;   // 8 waves (256 thr) per block
    gemm_wmma_k<K, FLAGS, HAS_BIAS><<<blocks, 256, 0, s>>>(A, lda, W, bias, resid,
                                                           ldr, out, ldo, M, NC);
}

// pad gat layer-0 weight [32,37] -> [32,40] with zero columns
__global__ void pad_w_k(const float* __restrict__ w, float* __restrict__ wpad)
{
    int i = blockIdx.x * blockDim.x + threadIdx.x;   // 32*40
    if (i >= 32 * 40) return;
    int n = i / 40, k = i % 40;
    wpad[i] = (k < 37) ? w[n * 37 + k] : 0.f;
}

// ---------------------------------------------------------------------------
// Node embedding gather -> node[N,40] (cols 37..39 zero-pad for K alignment)
// ---------------------------------------------------------------------------
__global__ void embed_k(const int* __restrict__ xi,
                        const float* __restrict__ t0, const float* __restrict__ t1,
                        const float* __restrict__ t2, const float* __restrict__ t3,
                        const float* __restrict__ t4, const float* __restrict__ t5,
                        const float* __restrict__ t6, const float* __restrict__ t7,
                        const float* __restrict__ t8, const float* __restrict__ t9,
                        float* __restrict__ node, int N)
{
    int i = blockIdx.x * blockDim.x + threadIdx.x;
    if (i >= N) return;
    const int* r = xi + (size_t)i * 10;
    float* o = node + (size_t)i * 40;
    const float* p;
    p = t0 + r[0]*12;
    #pragma unroll
    for (int j = 0; j < 12; ++j) o[j] = p[j];
    p = t1 + r[1]*3; o[12]=p[0]; o[13]=p[1]; o[14]=p[2];
    p = t2 + r[2]*3; o[15]=p[0]; o[16]=p[1]; o[17]=p[2];
    p = t3 + r[3]*3; o[18]=p[0]; o[19]=p[1]; o[20]=p[2];
    p = t4 + r[4]*3; o[21]=p[0]; o[22]=p[1]; o[23]=p[2];
    p = t5 + r[5]*3; o[24]=p[0]; o[25]=p[1]; o[26]=p[2];
    p = t6 + r[6]*3; o[27]=p[0]; o[28]=p[1]; o[29]=p[2];
    p = t7 + r[7]*3; o[30]=p[0]; o[31]=p[1]; o[32]=p[2];
    p = t8 + r[8]*2; o[33]=p[0]; o[34]=p[1];
    p = t9 + r[9]*2; o[35]=p[0]; o[36]=p[1];
    o[37] = 0.f; o[38] = 0.f; o[39] = 0.f;
}

__global__ void zero_k(float* __restrict__ p, int n)
{
    int i = blockIdx.x * blockDim.x + threadIdx.x;
    if (i < n) p[i] = 0.f;
}

// BatchNorm stats: per-block LDS partial sums then global atomic accumulate
#define BN_ROWS 2048
__global__ void bn_stats_k(const float* __restrict__ node, float* __restrict__ stats, int N)
{
    __shared__ float ssum[40];
    __shared__ float ssq[40];
    if (threadIdx.x < 40) { ssum[threadIdx.x] = 0.f; ssq[threadIdx.x] = 0.f; }
    __syncthreads();
    int r0 = blockIdx.x * BN_ROWS;
    int rows = N - r0; if (rows > BN_ROWS) rows = BN_ROWS;
    if (rows > 0) {
        int total = rows * 40;
        for (int t = threadIdx.x; t < total; t += blockDim.x) {
            int d = t % 40;
            if (d < 37) {
                float v = node[(size_t)r0 * 40 + t];
                atomicAdd(&ssum[d], v);
                atomicAdd(&ssq[d], v * v);
            }
        }
    }
    __syncthreads();
    if (threadIdx.x < 37) {
        atomicAdd(&stats[threadIdx.x],      ssum[threadIdx.x]);
        atomicAdd(&stats[40 + threadIdx.x], ssq[threadIdx.x]);
    }
}

__global__ void bn_apply_k(float* __restrict__ node, const float* __restrict__ stats,
                           const float* __restrict__ g, const float* __restrict__ b, int N)
{
    int i = blockIdx.x * blockDim.x + threadIdx.x;
    if (i >= N * 40) return;
    int d = i % 40;
    if (d >= 37) return;  // keep zero pad
    float invN = 1.f / (float)N;
    float m = stats[d] * invN;
    float var = fmaxf(stats[40 + d] * invN - m * m, 0.f);
    node[i] = (node[i] - m) * rsqrtf(var + 1e-5f) * g[d] + b[d];
}

// per-node att dots: as[i] = h[i,:].att_src ; ad[i] = h[i,:].att_dst (wave reduce)
__global__ void gat_dots_k(const float* __restrict__ h, const float* __restrict__ ws_,
                           const float* __restrict__ wd_, float* __restrict__ asrc,
                           float* __restrict__ adst, int N)
{
    int wid  = blockIdx.x * (blockDim.x >> 5) + (threadIdx.x >> 5);
    int lane = threadIdx.x & 31;
    if (wid >= N) return;
    float hv = h[(size_t)wid * 32 + lane];
    float ps = hv * ws_[lane];
    float pd = hv * wd_[lane];
    #pragma unroll
    for (int o = 16; o; o >>= 1) { ps += __shfl_xor(ps, o); pd += __shfl_xor(pd, o); }
    if (lane == 0) { asrc[wid] = ps; adst[wid] = pd; }
}

__global__ void gat_init_k(float* __restrict__ amax, float* __restrict__ denom,
                           float* __restrict__ agg, const float* __restrict__ bias, int N)
{
    int i = blockIdx.x * blockDim.x + threadIdx.x;
    if (i < N) { amax[i] = -INFINITY; denom[i] = 0.f; }
    if (i < N * 32) agg[i] = bias[i & 31];
}

__device__ __forceinline__ void atomicMaxFloat(float* addr, float v)
{
    if (v >= 0.f) atomicMax((int*)addr, __float_as_int(v));
    else          atomicMin((unsigned int*)addr, __float_as_uint(v));
}

__global__ void gat_edge1_k(const int* __restrict__ src, const int* __restrict__ dst,
                            const float* __restrict__ asrc, const float* __restrict__ adst,
                            float* __restrict__ ea, float* __restrict__ amax, int E)
{
    int e = blockIdx.x * blockDim.x + threadIdx.x;
    if (e >= E) return;
    float a = asrc[src[e]] + adst[dst[e]];
    a = (a > 0.f) ? a : 0.2f * a;           // leaky_relu(0.2)
    ea[e] = a;
    atomicMaxFloat(&amax[dst[e]], a);
}

__global__ void gat_edge2_k(const int* __restrict__ dst, float* __restrict__ ea,
                            const float* __restrict__ amax, float* __restrict__ denom, int E)
{
    int e = blockIdx.x * blockDim.x + threadIdx.x;
    if (e >= E) return;
    int d = dst[e];
    float ex = __expf(ea[e] - amax[d]);
    ea[e] = ex;
    atomicAdd(&denom[d], ex);
}

// one wave per edge, lane = channel: agg[dst,c] += alpha * h[src,c]
__global__ void gat_edge3_k(const int* __restrict__ src, const int* __restrict__ dst,
                            const float* __restrict__ ea, const float* __restrict__ denom,
                            const float* __restrict__ h, float* __restrict__ agg, int E)
{
    int wid  = blockIdx.x * (blockDim.x >> 5) + (threadIdx.x >> 5);
    int lane = threadIdx.x & 31;
    if (wid >= E) return;
    int s = src[wid], d = dst[wid];
    float alpha = ea[wid] / (denom[d] + 1e-16f);
    atomicAdd(&agg[(size_t)d * 32 + lane], alpha * h[(size_t)s * 32 + lane]);
}

__global__ void relu_k(float* __restrict__ x, int n)
{
    int i = blockIdx.x * blockDim.x + threadIdx.x;
    if (i < n) x[i] = fmaxf(x[i], 0.f);
}

// ---------------------------------------------------------------------------
// Encoder self-attention: one wave per graph (25 tokens, 4 heads, d=8).
// lane = channel = head*8 + d. qkv[N,96]: q@0, k@32, v@64.
// ---------------------------------------------------------------------------
__global__ void attn_enc_k(const float* __restrict__ qkv, float* __restrict__ out, int B)
{
    __shared__ float kbuf[4][25][32];
    __shared__ float vbuf[4][25][32];
    int w    = threadIdx.x >> 5;
    int g    = blockIdx.x * (blockDim.x >> 5) + w;
    int lane = threadIdx.x & 31;
    if (g >= B) return;
    const float* base = qkv + (size_t)g * 25 * 96;
    #pragma unroll
    for (int j = 0; j < 25; ++j) {
        kbuf[w][j][lane] = base[j * 96 + 32 + lane];
        vbuf[w][j][lane] = base[j * 96 + 64 + lane];
    }
    const float scale = 0.35355339059327373f;  // 8^-0.5
    for (int i = 0; i < 25; ++i) {
        float q = base[i * 96 + lane];
        float s[25];
        #pragma unroll
        for (int j = 0; j < 25; ++j) {
            float p = q * kbuf[w][j][lane];
            p += __shfl_xor(p, 1); p += __shfl_xor(p, 2); p += __shfl_xor(p, 4);
            s[j] = p * scale;                  // identical across the head's 8 lanes
        }
        float mx = s[0];
        #pragma unroll
        for (int j = 1; j < 25; ++j) mx = fmaxf(mx, s[j]);
        float den = 0.f;
        #pragma unroll
        for (int j = 0; j < 25; ++j) { s[j] = __expf(s[j] - mx); den += s[j]; }
        float inv = 1.f / den;
        float o = 0.f;
        #pragma unroll
        for (int j = 0; j < 25; ++j) o += s[j] * vbuf[w][j][lane];
        out[((size_t)g * 25 + i) * 32 + lane] = o * inv;
    }
}

// q_seed[32] = seed @ wq^T + bq  (tiny, one wave)
__global__ void seed_q_k(const float* __restrict__ seed, const float* __restrict__ in_w,
                         const float* __restrict__ in_b, float* __restrict__ qseed)
{
    int c = threadIdx.x;
    float acc = in_b[c];
    #pragma unroll
    for (int k = 0; k < 32; ++k) acc += seed[k] * in_w[c * 32 + k];
    qseed[c] = acc;
}

// PMA attention: 1 query (seed), 25 keys. kv[N,64]: k@0, v@32. one wave/graph.
__global__ void attn_pma_k(const float* __restrict__ kv, const float* __restrict__ qseed,
                           float* __restrict__ out, int B)
{
    int g    = blockIdx.x * (blockDim.x >> 5) + (threadIdx.x >> 5);
    int lane = threadIdx.x & 31;
    if (g >= B) return;
    const float scale = 0.35355339059327373f;
    float q = qseed[lane];
    const float* base = kv + (size_t)g * 25 * 64;
    float s[25];
    #pragma unroll
    for (int j = 0; j < 25; ++j) {
        float p = q * base[j * 64 + lane];
        p += __shfl_xor(p, 1); p += __shfl_xor(p, 2); p += __shfl_xor(p, 4);
        s[j] = p * scale;
    }
    float mx = s[0];
    #pragma unroll
    for (int j = 1; j < 25; ++j) mx = fmaxf(mx, s[j]);
    float den = 0.f;
    #pragma unroll
    for (int j = 0; j < 25; ++j) { s[j] = __expf(s[j] - mx); den += s[j]; }
    float inv = 1.f / den;
    float o = 0.f;
    #pragma unroll
    for (int j = 0; j < 25; ++j) o += s[j] * base[j * 64 + 32 + lane];
    out[(size_t)g * 32 + lane] = o * inv;
}

__global__ void finalize_k(const float* __restrict__ x, const int* __restrict__ valid,
                           float* __restrict__ out, int B)
{
    int i = blockIdx.x * blockDim.x + threadIdx.x;
    int n = B * 32;
    if (i < n) {
        float v = x[i];
        if (isnan(v)) v = 0.f;
        else if (isinf(v)) v = (v > 0.f) ? 3.4028234663852886e38f : -3.4028234663852886e38f;
        out[i] = v;
    }
    if (i < B) out[n + i] = (float)valid[i];
}

// ---------------------------------------------------------------------------
extern "C" void kernel_launch(void* const* d_in, const int* in_sizes, int n_in,
                              void* d_out, int out_size, void* d_ws, size_t ws_size,
                              hipStream_t stream)
{
    (void)n_in; (void)out_size; (void)ws_size;
    const int N = in_sizes[0] / 10;
    const int E = in_sizes[2] / 2;
    const int B = in_sizes[4];

    auto F = [&](int i) { return (const float*)d_in[i]; };
    const int* x_idx = (const int*)d_in[0];
    const int* src   = (const int*)d_in[2];
    const int* dst   = src + E;
    const int* valid = (const int*)d_in[4];

    // d_in map (setup_inputs insertion order, recursively flattened):
    // 0 x_idx, 1 edge_attr_idx, 2 edge_index, 3 batch, 4 smiles_valid,
    // 5..14 node_tables, 15..19 edge_tables, 20,21 bn_node g/b, 22,23 bn_edge g/b,
    // 24..43 gat{w,att_src,att_dst,bias}x5, 44..67 enc mab x4 {in_w,in_b,out_w,out_b,lin_w,lin_b},
    // 68 pma_lin_w, 69 pma_lin_b, 70 seed, 71..76 pma_mab, 77..100 dec mab x4.

    // workspace layout (floats); node[N,40] aliases qkv[N,96] (disjoint lifetimes)
    float* ws    = (float*)d_ws;
    float* node  = ws;
    float* qkv   = ws;
    float* bufA  = ws   + (size_t)N * 96;
    float* bufB  = bufA + (size_t)N * 32;
    float* bufC  = bufB + (size_t)N * 32;
    float* asrc  = bufC + (size_t)N * 32;
    float* adst  = asrc + N;
    float* amax  = adst + N;
    float* denom = amax + N;
    float* ea    = denom + N;
    float* stats = ea + E;       // 128
    float* qseed = stats + 128;  // 32
    float* wpad  = qseed + 32;   // 32*40 padded layer-0 GAT weight

    // 1) embed + batchnorm
    zero_k<<<1, 128, 0, stream>>>(stats, 128);
    embed_k<<<(N + 255) / 256, 256, 0, stream>>>(x_idx, F(5), F(6), F(7), F(8), F(9),
                                                 F(10), F(11), F(12), F(13), F(14), node, N);
    bn_stats_k<<<(N + BN_ROWS - 1) / BN_ROWS, 256, 0, stream>>>(node, stats, N);
    bn_apply_k<<<(N * 40 + 255) / 256, 256, 0, stream>>>(node, stats, F(20), F(21), N);

    // 2) five GAT layers (ping-pong bufA/bufC for aggregated output; H in bufB)
    pad_w_k<<<(32 * 40 + 127) / 128, 128, 0, stream>>>(F(24), wpad);
    const float* X = bufA;  // set per layer below
    float* pool[2] = { bufA, bufC };
    int pi = 0;
    for (int i = 0; i < 5; ++i) {
        const float* aw  = F(25 + 4 * i);
        const float* adw = F(26 + 4 * i);
        const float* bv  = F(27 + 4 * i);
        float* H   = bufB;
        float* agg = pool[pi];
        if (i == 0) {
            launch_gemm<40, 0, false>(node, 40, wpad, nullptr, nullptr, 0,
                                      H, 32, N, 32, stream);
        } else {
            launch_gemm<32, 0, false>(X, 32, F(24 + 4 * i), nullptr, nullptr, 0,
                                      H, 32, N, 32, stream);
        }
        gat_dots_k<<<(N + 7) / 8, 256, 0, stream>>>(H, aw, adw, asrc, adst, N);
        gat_init_k<<<(N * 32 + 255) / 256, 256, 0, stream>>>(amax, denom, agg, bv, N);
        gat_edge1_k<<<(E + 255) / 256, 256, 0, stream>>>(src, dst, asrc, adst, ea, amax, E);
        gat_edge2_k<<<(E + 255) / 256, 256, 0, stream>>>(dst, ea, amax, denom, E);
        gat_edge3_k<<<(E + 7) / 8, 256, 0, stream>>>(src, dst, ea, denom, H, agg, E);
        if (i < 4) relu_k<<<(N * 32 + 255) / 256, 256, 0, stream>>>(agg, N * 32);
        X = agg; pi ^= 1;
    }
    // final node states in bufA (layer 4 wrote pool[0])

    // 3) four encoder MABs on [B,25,32] (== [N,32], batch is a pure reshape)
    for (int l = 0; l < 4; ++l) {
        const float* in_w  = F(44 + 6 * l + 0);
        const float* in_b  = F(44 + 6 * l + 1);
        const float* out_w = F(44 + 6 * l + 2);
        const float* out_b = F(44 + 6 * l + 3);
        const float* lin_w = F(44 + 6 * l + 4);
        const float* lin_b = F(44 + 6 * l + 5);
        launch_gemm<32, 0, true>(bufA, 32, in_w, in_b, nullptr, 0, qkv, 96, N, 96, stream);
        attn_enc_k<<<(B + 3) / 4, 128, 0, stream>>>(qkv, bufB, B);
        launch_gemm<32, GF_ADD_RESID, true>(bufB, 32, out_w, out_b, bufA, 32,
                                            bufC, 32, N, 32, stream);
        launch_gemm<32, GF_RELU_ADD, true>(bufC, 32, lin_w, lin_b, bufC, 32,
                                           bufA, 32, N, 32, stream);
    }

    // 4) PMA: z = relu(x @ pma_lin), then MAB(seed, z)
    launch_gemm<32, GF_RELU, true>(bufA, 32, F(68), F(69), nullptr, 0,
                                   bufB, 32, N, 32, stream);                   // z
    const float* p_in_w  = F(71); const float* p_in_b  = F(72);
    const float* p_out_w = F(73); const float* p_out_b = F(74);
    const float* p_lin_w = F(75); const float* p_lin_b = F(76);
    launch_gemm<32, 0, true>(bufB, 32, p_in_w + 32 * 32, p_in_b + 32, nullptr, 0,
                             qkv, 64, N, 64, stream);                          // k,v
    seed_q_k<<<1, 32, 0, stream>>>(F(70), p_in_w, p_in_b, qseed);
    attn_pma_k<<<(B + 7) / 8, 256, 0, stream>>>(qkv, qseed, bufC, B);
    launch_gemm<32, GF_ADD_RESID | GF_RESID_VEC, true>(bufC, 32, p_out_w, p_out_b,
                                                       F(70), 0, bufB, 32, B, 32,
                                                       stream);               // u = h+seed
    launch_gemm<32, GF_RELU_ADD, true>(bufB, 32, p_lin_w, p_lin_b, bufB, 32,
                                       bufA, 32, B, 32, stream);              // x

    // 5) four decoder MABs on [B,1,32]: softmax over 1 key == identity -> MHA = v-path
    for (int l = 0; l < 4; ++l) {
        const float* in_w  = F(77 + 6 * l + 0);
        const float* in_b  = F(77 + 6 * l + 1);
        const float* out_w = F(77 + 6 * l + 2);
        const float* out_b = F(77 + 6 * l + 3);
        const float* lin_w = F(77 + 6 * l + 4);
        const float* lin_b = F(77 + 6 * l + 5);
        launch_gemm<32, 0, true>(bufA, 32, in_w + 64 * 32, in_b + 64, nullptr, 0,
                                 bufB, 32, B, 32, stream);                     // v
        launch_gemm<32, GF_ADD_RESID, true>(bufB, 32, out_w, out_b, bufA, 32,
                                            bufC, 32, B, 32, stream);          // u
        launch_gemm<32, GF_RELU_ADD, true>(bufC, 32, lin_w, lin_b, bufC, 32,
                                           bufA, 32, B, 32, stream);           // x
    }

    // 6) g = nan_to_num(x), mask = float(smiles_valid)
    finalize_k<<<(B * 32 + 255) / 256, 256, 0, stream>>>(bufA, valid, (float*)d_out, B);
}